// TrajNet_14663018349172
// MI455X (gfx1250) — compile-verified
//
#include <hip/hip_runtime.h>

typedef __attribute__((ext_vector_type(2))) float v2f;
typedef __attribute__((ext_vector_type(8))) float v8f;

// workspace layout (float offsets)
#define WS_P     0       // [32][16]  P padded (B-frags for S@P)
#define WS_PT    512     // [12][32]  P^T padded (B-frags for sol@P^T)
#define WS_CINV  896     // [12][16]  cost^-1 padded
#define WS_H     1088    // [12][16]  4*rho_obs*P^T P + rho_eq*Aeq^T Aeq
#define WS_AEQ   1280    // [8][16]   A_eq padded
#define WS_CC    1408    // [2][16]   rho_obs*S{x,y}*colsum(P)
#define WS_COLP  1440    // [16]      colsum(P)
#define WS_BGEN  1472    // [1024][16] b_gen padded to 16 cols

#define MEMBAR() asm volatile("" ::: "memory")

static __device__ __forceinline__ v8f wmma4(v2f a, v2f b, v8f c) {
  return __builtin_amdgcn_wmma_f32_16x16x4_f32(false, a, false, b, (short)0, c,
                                               false, false);
}
static __device__ __forceinline__ v8f splat8(float s) {
  v8f v = {s, s, s, s, s, s, s, s};
  return v;
}

// ---------------- setup: rebuild reference constants on device ----------------
static __device__ double dcomb(int n, int k) {
  if (k < 0 || k > n) return 0.0;
  double r = 1.0;
  for (int i = 1; i <= k; ++i) r = r * (double)(n - k + i) / (double)i;
  return r;
}
static __device__ double dpow(double x, int e) {
  double r = 1.0;
  for (int i = 0; i < e; ++i) r *= x;
  return r;
}
static __device__ double bern(int n, int k, double t) {
  if (k < 0 || k > n) return 0.0;
  return dcomb(n, k) * dpow(t, k) * dpow(1.0 - t, n - k);
}

__global__ void setup_kernel(float* ws) {
  if (threadIdx.x != 0 || blockIdx.x != 0) return;
  double P[30][11], Pdd[30][11];
  for (int t = 0; t < 30; ++t) {
    double tv = (double)t / 29.0;
    for (int k = 0; k < 11; ++k) {
      P[t][k] = bern(10, k, tv);
      Pdd[t][k] = 90.0 * (bern(8, k - 2, tv) - 2.0 * bern(8, k - 1, tv) + bern(8, k, tv));
    }
  }
  double AEQ[6][11];
  for (int k = 0; k < 11; ++k) {
    AEQ[0][k] = P[0][k];
    AEQ[1][k] = 10.0 * (bern(9, k - 1, 0.0) - bern(9, k, 0.0));
    AEQ[2][k] = Pdd[0][k];
    AEQ[3][k] = P[29][k];
    AEQ[4][k] = 10.0 * (bern(9, k - 1, 1.0) - bern(9, k, 1.0));
    AEQ[5][k] = Pdd[29][k];
  }
  double M[11][11], Q[11][11], G[11][11];
  for (int i = 0; i < 11; ++i)
    for (int j = 0; j < 11; ++j) {
      double m = 0, q = 0, g = 0;
      for (int t = 0; t < 30; ++t) { m += P[t][i] * P[t][j]; q += Pdd[t][i] * Pdd[t][j]; }
      for (int r = 0; r < 6; ++r) g += AEQ[r][i] * AEQ[r][j];
      M[i][j] = m; Q[i][j] = q; G[i][j] = g;
    }
  // cost = 10*Q + 1.2*4*M + 10*G ; invert via Gauss-Jordan (SPD, pivoted)
  double A[11][22];
  for (int i = 0; i < 11; ++i)
    for (int j = 0; j < 11; ++j) {
      A[i][j] = 10.0 * Q[i][j] + 4.8 * M[i][j] + 10.0 * G[i][j];
      A[i][11 + j] = (i == j) ? 1.0 : 0.0;
    }
  for (int col = 0; col < 11; ++col) {
    int piv = col;
    double best = fabs(A[col][col]);
    for (int r = col + 1; r < 11; ++r) {
      double v = fabs(A[r][col]);
      if (v > best) { best = v; piv = r; }
    }
    if (piv != col)
      for (int j = 0; j < 22; ++j) { double tmp = A[col][j]; A[col][j] = A[piv][j]; A[piv][j] = tmp; }
    double pv = A[col][col];
    for (int j = 0; j < 22; ++j) A[col][j] /= pv;
    for (int r = 0; r < 11; ++r)
      if (r != col) {
        double f = A[r][col];
        for (int j = 0; j < 22; ++j) A[r][j] -= f * A[col][j];
      }
  }
  for (int r = 0; r < 32; ++r)
    for (int c = 0; c < 16; ++c)
      ws[WS_P + r * 16 + c] = (r < 30 && c < 11) ? (float)P[r][c] : 0.f;
  for (int r = 0; r < 12; ++r)
    for (int c = 0; c < 32; ++c)
      ws[WS_PT + r * 32 + c] = (r < 11 && c < 30) ? (float)P[c][r] : 0.f;
  for (int r = 0; r < 12; ++r)
    for (int c = 0; c < 16; ++c)
      ws[WS_CINV + r * 16 + c] = (r < 11 && c < 11) ? (float)A[r][11 + c] : 0.f;
  for (int r = 0; r < 12; ++r)
    for (int c = 0; c < 16; ++c)
      ws[WS_H + r * 16 + c] = (r < 11 && c < 11) ? (float)(4.8 * M[r][c] + 10.0 * G[r][c]) : 0.f;
  for (int r = 0; r < 8; ++r)
    for (int c = 0; c < 16; ++c)
      ws[WS_AEQ + r * 16 + c] = (r < 6 && c < 11) ? (float)AEQ[r][c] : 0.f;
  double Sx = -10000.0 + 10000.79 + 30000.0 + 10000.0;
  double Sy = -10000.0 + 10000.0 - 30000.8 + 10000.0;
  for (int c = 0; c < 16; ++c) {
    double cp = 0.0;
    if (c < 11)
      for (int t = 0; t < 30; ++t) cp += P[t][c];
    ws[WS_CC + c] = (float)(1.2 * Sx * cp);
    ws[WS_CC + 16 + c] = (float)(1.2 * Sy * cp);
    ws[WS_COLP + c] = (float)cp;
  }
}

// ---------------- MLP: b_gen = mask*b + (1-mask)*(relu(xW1+b1)W2+b2) ----------
// All OOB guards implemented as clamp-address + select (no exec-mask branches).
__global__ __launch_bounds__(32) void mlp_kernel(const float* __restrict__ x,
                                                 const float* __restrict__ bin,
                                                 const float* __restrict__ W1,
                                                 const float* __restrict__ b1,
                                                 const float* __restrict__ W2,
                                                 const float* __restrict__ b2,
                                                 float* __restrict__ bgen) {
  __shared__ float hbuf[16 * 128];
  const int l = threadIdx.x;
  const int c = l & 15;
  const int koff = (l >> 4) * 2;
  const int rowbase = 8 * (l >> 4);
  const int r0 = blockIdx.x * 16;
  const float* xrow = x + (r0 + c) * 54;

  v8f acc[8];
#pragma unroll
  for (int nt = 0; nt < 8; ++nt) acc[nt] = splat8(0.f);

  // h = x @ W1 : k = 0..12 -> rows 4k+koff+1 <= 53, always in range
#pragma unroll
  for (int k = 0; k < 13; ++k) {
    const int ka = 4 * k + koff;
    v2f a;
    a.x = xrow[ka];
    a.y = xrow[ka + 1];
#pragma unroll
    for (int nt = 0; nt < 8; ++nt) {
      v2f bf;
      bf.x = W1[ka * 128 + nt * 16 + c];
      bf.y = W1[(ka + 1) * 128 + nt * 16 + c];
      acc[nt] = wmma4(a, bf, acc[nt]);
    }
  }
  // k = 13 tail: rows 52,53 valid (koff==0); rows 54,55 zero (koff==2)
  {
    const bool vld = (koff == 0);
    const int ka = vld ? 52 : 0;  // clamped, always-valid address
    v2f a;
    float ax = xrow[ka];
    float ay = xrow[ka + 1];
    a.x = vld ? ax : 0.f;
    a.y = vld ? ay : 0.f;
#pragma unroll
    for (int nt = 0; nt < 8; ++nt) {
      float b0 = W1[ka * 128 + nt * 16 + c];
      float b1v = W1[(ka + 1) * 128 + nt * 16 + c];
      v2f bf;
      bf.x = vld ? b0 : 0.f;
      bf.y = vld ? b1v : 0.f;
      acc[nt] = wmma4(a, bf, acc[nt]);
    }
  }
  // + b1, relu, stage to LDS
#pragma unroll
  for (int nt = 0; nt < 8; ++nt) {
    float bias = b1[nt * 16 + c];
#pragma unroll
    for (int i = 0; i < 8; ++i) {
      float h = acc[nt][i] + bias;
      hbuf[(i + rowbase) * 128 + nt * 16 + c] = h > 0.f ? h : 0.f;
    }
  }
  MEMBAR();
  // b_pred = h @ W2  (K=128); columns >= 12 forced to zero via select
  const bool cvld = (c < 12);
  const int ccl = cvld ? c : 0;  // clamped column, always-valid address
  v8f accP = splat8(0.f);
#pragma unroll
  for (int k = 0; k < 32; ++k) {
    v2f a2 = *(const v2f*)&hbuf[c * 128 + 4 * k + koff];
    const int rw = 4 * k + koff;
    float w0 = W2[rw * 12 + ccl];
    float w1v = W2[(rw + 1) * 12 + ccl];
    v2f bw;
    bw.x = cvld ? w0 : 0.f;
    bw.y = cvld ? w1v : 0.f;
    accP = wmma4(a2, bw, accP);
  }
  float bias2 = b2[ccl];
#pragma unroll
  for (int i = 0; i < 8; ++i) {
    const int g = r0 + i + rowbase;
    float braw = bin[g * 12 + ccl];          // unconditional, clamped address
    float pred = accP[i] + bias2;
    float val = (c == 3 || c == 9) ? pred : braw;
    bgen[g * 16 + c] = cvld ? val : 0.f;
  }
}

// ---------------- ADMM: 300 iterations, one wave = 16 problems ----------------
__global__ __launch_bounds__(32) void admm_kernel(const float* __restrict__ ws,
                                                  float* __restrict__ out) {
  __shared__ float bufT[16 * 16];
  __shared__ float xpb[16 * 32];
  __shared__ float ypb[16 * 32];
  __shared__ float sdc[16 * 32];
  __shared__ float sds[16 * 32];

  const int l = threadIdx.x;
  const int c = l & 15;
  const int koff = (l >> 4) * 2;
  const int rowbase = 8 * (l >> 4);
  const int r0 = blockIdx.x * 16;

  // constant B-fragments (row 4s+koff / +1, col c)
  v2f CINVb[3], Hb[3], Pb[8], PTb[3][2], AEQb[2];
#pragma unroll
  for (int s = 0; s < 3; ++s) {
    CINVb[s].x = ws[WS_CINV + (4 * s + koff) * 16 + c];
    CINVb[s].y = ws[WS_CINV + (4 * s + koff + 1) * 16 + c];
    Hb[s].x = ws[WS_H + (4 * s + koff) * 16 + c];
    Hb[s].y = ws[WS_H + (4 * s + koff + 1) * 16 + c];
#pragma unroll
    for (int nt = 0; nt < 2; ++nt) {
      PTb[s][nt].x = ws[WS_PT + (4 * s + koff) * 32 + nt * 16 + c];
      PTb[s][nt].y = ws[WS_PT + (4 * s + koff + 1) * 32 + nt * 16 + c];
    }
  }
#pragma unroll
  for (int s = 0; s < 8; ++s) {
    Pb[s].x = ws[WS_P + (4 * s + koff) * 16 + c];
    Pb[s].y = ws[WS_P + (4 * s + koff + 1) * 16 + c];
  }
#pragma unroll
  for (int s = 0; s < 2; ++s) {
    AEQb[s].x = ws[WS_AEQ + (4 * s + koff) * 16 + c];
    AEQb[s].y = ws[WS_AEQ + (4 * s + koff + 1) * 16 + c];
  }
  const float ccx = ws[WS_CC + c];
  const float ccy = ws[WS_CC + 16 + c];
  const float colp = ws[WS_COLP + c];

  // Cx = rho_eq*(bx@Aeq) + rho_obs*Sx*colP   (per-batch, loop-invariant)
  const float* bgen = ws + WS_BGEN;
  const int grow = r0 + c;
  v8f EBX = splat8(0.f), EBY = splat8(0.f);
#pragma unroll
  for (int s = 0; s < 2; ++s) {
    v2f ax = *(const v2f*)&bgen[grow * 16 + 4 * s + koff];
    EBX = wmma4(ax, AEQb[s], EBX);
    v2f ay = *(const v2f*)&bgen[grow * 16 + 6 + 4 * s + koff];
    EBY = wmma4(ay, AEQb[s], EBY);
  }
  v8f Cx = EBX * 10.0f + ccx;
  v8f Cy = EBY * 10.0f + ccy;

  v8f lamx = splat8(0.f), lamy = splat8(0.f);
  v8f Vx = splat8(4.0f * colp);  // d=1, alpha=0 -> sumDC = 4
  v8f Vy = splat8(0.f);

  // zero pad-columns 30,31 of sdc/sds
  {
    int rr = l >> 1, cc2 = 30 + (l & 1);
    sdc[rr * 32 + cc2] = 0.f;
    sds[rr * 32 + cc2] = 0.f;
  }
  MEMBAR();

  const float xcs[4] = {-10000.0f, 10000.79f, 30000.0f, 10000.0f};
  const float ycs[4] = {-10000.0f, 10000.0f, -30000.8f, 10000.0f};

#pragma unroll 1
  for (int it = 0; it < 300; ++it) {
    v8f HX, HY;
    // -------- X side --------
    {
      v8f T = lamx + Vx * 1.2f + Cx;
#pragma unroll
      for (int i = 0; i < 8; ++i) bufT[(i + rowbase) * 16 + c] = T[i];
      MEMBAR();
      v2f a0 = *(const v2f*)&bufT[c * 16 + koff];
      v2f a1 = *(const v2f*)&bufT[c * 16 + 4 + koff];
      v2f a2 = *(const v2f*)&bufT[c * 16 + 8 + koff];
      v8f sol = splat8(0.f);
      sol = wmma4(a0, CINVb[0], sol);
      sol = wmma4(a1, CINVb[1], sol);
      sol = wmma4(a2, CINVb[2], sol);
      MEMBAR();
#pragma unroll
      for (int i = 0; i < 8; ++i) bufT[(i + rowbase) * 16 + c] = sol[i];
      MEMBAR();
      v2f s0 = *(const v2f*)&bufT[c * 16 + koff];
      v2f s1 = *(const v2f*)&bufT[c * 16 + 4 + koff];
      v2f s2 = *(const v2f*)&bufT[c * 16 + 8 + koff];
      v8f xp0 = splat8(0.f), xp1 = splat8(0.f);
      xp0 = wmma4(s0, PTb[0][0], xp0);
      xp0 = wmma4(s1, PTb[1][0], xp0);
      xp0 = wmma4(s2, PTb[2][0], xp0);
      xp1 = wmma4(s0, PTb[0][1], xp1);
      xp1 = wmma4(s1, PTb[1][1], xp1);
      xp1 = wmma4(s2, PTb[2][1], xp1);
      HX = splat8(0.f);
      HX = wmma4(s0, Hb[0], HX);
      HX = wmma4(s1, Hb[1], HX);
      HX = wmma4(s2, Hb[2], HX);
#pragma unroll
      for (int i = 0; i < 8; ++i) {
        xpb[(i + rowbase) * 32 + c] = xp0[i];
        xpb[(i + rowbase) * 32 + 16 + c] = xp1[i];
      }
    }
    // -------- Y side --------
    {
      v8f T = lamy + Vy * 1.2f + Cy;
      MEMBAR();
#pragma unroll
      for (int i = 0; i < 8; ++i) bufT[(i + rowbase) * 16 + c] = T[i];
      MEMBAR();
      v2f a0 = *(const v2f*)&bufT[c * 16 + koff];
      v2f a1 = *(const v2f*)&bufT[c * 16 + 4 + koff];
      v2f a2 = *(const v2f*)&bufT[c * 16 + 8 + koff];
      v8f sol = splat8(0.f);
      sol = wmma4(a0, CINVb[0], sol);
      sol = wmma4(a1, CINVb[1], sol);
      sol = wmma4(a2, CINVb[2], sol);
      MEMBAR();
#pragma unroll
      for (int i = 0; i < 8; ++i) bufT[(i + rowbase) * 16 + c] = sol[i];
      MEMBAR();
      v2f s0 = *(const v2f*)&bufT[c * 16 + koff];
      v2f s1 = *(const v2f*)&bufT[c * 16 + 4 + koff];
      v2f s2 = *(const v2f*)&bufT[c * 16 + 8 + koff];
      v8f yp0 = splat8(0.f), yp1 = splat8(0.f);
      yp0 = wmma4(s0, PTb[0][0], yp0);
      yp0 = wmma4(s1, PTb[1][0], yp0);
      yp0 = wmma4(s2, PTb[2][0], yp0);
      yp1 = wmma4(s0, PTb[0][1], yp1);
      yp1 = wmma4(s1, PTb[1][1], yp1);
      yp1 = wmma4(s2, PTb[2][1], yp1);
      HY = splat8(0.f);
      HY = wmma4(s0, Hb[0], HY);
      HY = wmma4(s1, Hb[1], HY);
      HY = wmma4(s2, Hb[2], HY);
#pragma unroll
      for (int i = 0; i < 8; ++i) {
        ypb[(i + rowbase) * 32 + c] = yp0[i];
        ypb[(i + rowbase) * 32 + 16 + c] = yp1[i];
      }
    }
    MEMBAR();
    // -------- elementwise: sumDC/sumDS over 4 obstacles (no trig needed) -----
#pragma unroll 1
    for (int i2 = 0; i2 < 15; ++i2) {
      int idx = l + 32 * i2;
      int bb = idx / 30;
      int tt = idx - bb * 30;
      float px = xpb[bb * 32 + tt];
      float py = ypb[bb * 32 + tt];
      float s1s = 0.f, s2s = 0.f;
#pragma unroll
      for (int o = 0; o < 4; ++o) {
        float wcx = px - xcs[o];
        float wcy = py - ycs[o];
        float r2 = wcx * wcx + wcy * wcy;
        float rs = (r2 > 0.f) ? __builtin_amdgcn_rsqf(r2) : 0.f;
        float sc = (r2 >= 1.f) ? 1.f : rs;
        s1s += (r2 > 0.f) ? wcx * sc : 1.0f;  // d*cos(alpha)
        s2s += (r2 > 0.f) ? wcy * sc : 0.0f;  // d*sin(alpha)
      }
      sdc[bb * 32 + tt] = s1s;
      sds[bb * 32 + tt] = s2s;
    }
    MEMBAR();
    // -------- V = sumDC @ P  and lambda update ------------------------------
    Vx = splat8(0.f);
    Vy = splat8(0.f);
#pragma unroll
    for (int s = 0; s < 8; ++s) {
      v2f avx = *(const v2f*)&sdc[c * 32 + 4 * s + koff];
      Vx = wmma4(avx, Pb[s], Vx);
      v2f avy = *(const v2f*)&sds[c * 32 + 4 * s + koff];
      Vy = wmma4(avy, Pb[s], Vy);
    }
    lamx = lamx - HX + Vx * 1.2f + Cx;
    lamy = lamy - HY + Vy * 1.2f + Cy;
    MEMBAR();
  }

  // output: xpb/ypb hold final sol @ P^T
  MEMBAR();
#pragma unroll 1
  for (int i2 = 0; i2 < 15; ++i2) {
    int idx = l + 32 * i2;
    int bb = idx / 30;
    int tt = idx - bb * 30;
    out[(r0 + bb) * 60 + tt] = xpb[bb * 32 + tt];
    out[(r0 + bb) * 60 + 30 + tt] = ypb[bb * 32 + tt];
  }
}

extern "C" void kernel_launch(void* const* d_in, const int* in_sizes, int n_in,
                              void* d_out, int out_size, void* d_ws, size_t ws_size,
                              hipStream_t stream) {
  const float* x = (const float*)d_in[0];
  const float* b = (const float*)d_in[1];
  const float* W1 = (const float*)d_in[2];
  const float* b1 = (const float*)d_in[3];
  const float* W2 = (const float*)d_in[4];
  const float* b2 = (const float*)d_in[5];
  float* ws = (float*)d_ws;
  float* out = (float*)d_out;

  setup_kernel<<<1, 1, 0, stream>>>(ws);
  mlp_kernel<<<64, 32, 0, stream>>>(x, b, W1, b1, W2, b2, ws + WS_BGEN);
  admm_kernel<<<64, 32, 0, stream>>>(ws, out);
}